// KroneckerProduct_78486232367279
// MI455X (gfx1250) — compile-verified
//
#include <hip/hip_runtime.h>

typedef __attribute__((ext_vector_type(2))) float v2f;
typedef __attribute__((ext_vector_type(4))) float v4f;
typedef __attribute__((ext_vector_type(8))) float v8f;

// Problem dims (match reference)
#define AR 128
#define AC 128
#define BR 64
#define BC 64
// out: (AR*BR) x (AC*BC) = 8192 x 8192 f32

// One wave32 per output row r = i*BR + p.
// Row content (as j-major 128x64): M[j][q] = A[i,j] * B[p,q]  (rank-1 outer product).
// Each V_WMMA_F32_16X16X4_F32 computes a 16(q) x 16(j) outer-product tile:
//   Aop[m,k] = 0.25*B[p, q0+m]  (same for all k)  -> K-layout agnostic
//   Bop[k,n] = A[i, j0+n]       (same for all k)
//   D[m,n]   = sum_k 0.25*b*a = B[p,q0+m]*A[i,j0+n]
// D layout: lane l holds N = l%16, M = vgpr + 8*(l>=16)
// -> each lane's 8 D VGPRs are 8 *consecutive* floats in the output row,
//    stored as two non-temporal b128 stores.
__global__ __launch_bounds__(256) void kron_wmma_kernel(
    const float* __restrict__ A, const float* __restrict__ B,
    float* __restrict__ out) {
  const int lane = threadIdx.x & 31;
  const int wave = threadIdx.x >> 5;
  const int r    = blockIdx.x * 8 + wave;   // output row, 0..8191 (grid exact)
  const int i    = r >> 6;                  // A row
  const int p    = r & (BR - 1);            // B row
  const int half = lane >> 4;               // 0: lanes 0-15, 1: lanes 16-31
  const int ln   = lane & 15;

  const float* __restrict__ Arow = A + i * AC;
  const float* __restrict__ Brow = B + p * BC;
  float* __restrict__ rowOut = out + (size_t)r * (size_t)(AC * BC);

  for (int q0 = 0; q0 < BC; q0 += 16) {
    // A-operand: value depends only on M = ln; both K-VGPRs identical.
    const float bval = 0.25f * Brow[q0 + ln];
    v2f aop;
    aop.x = bval;
    aop.y = bval;

#pragma unroll
    for (int j0 = 0; j0 < AC; j0 += 16) {
      // B-operand: value depends only on N = ln; both K-VGPRs identical.
      const float aval = Arow[j0 + ln];
      v2f bop;
      bop.x = aval;
      bop.y = aval;

      v8f c = {};
      // 8 args: (neg_a, A, neg_b, B, c_mod, C, reuse_a, reuse_b)
      v8f d = __builtin_amdgcn_wmma_f32_16x16x4_f32(
          /*neg_a=*/false, aop, /*neg_b=*/false, bop,
          /*c_mod=*/(short)0, c, /*reuse_a=*/false, /*reuse_b=*/false);

      // Per-lane contiguous 8 floats: (j0+ln)*BC + q0 + 8*half + m'
      float* base = rowOut + (j0 + ln) * BC + q0 + 8 * half;
      v4f lo = __builtin_shufflevector(d, d, 0, 1, 2, 3);
      v4f hi = __builtin_shufflevector(d, d, 4, 5, 6, 7);
      __builtin_nontemporal_store(lo, (v4f*)base);        // global_store_b128 (NT)
      __builtin_nontemporal_store(hi, (v4f*)(base + 4));  // global_store_b128 (NT)
    }
  }
}

extern "C" void kernel_launch(void* const* d_in, const int* in_sizes, int n_in,
                              void* d_out, int out_size, void* d_ws, size_t ws_size,
                              hipStream_t stream) {
  const float* A = (const float*)d_in[0];  // 128*128 f32
  const float* B = (const float*)d_in[1];  // 64*64 f32
  float* out = (float*)d_out;              // 8192*8192 f32

  // 8192 output rows, one wave32 per row, 8 waves (256 threads) per block.
  dim3 grid((AR * BR) / 8);
  dim3 block(256);
  hipLaunchKernelGGL(kron_wmma_kernel, grid, block, 0, stream, A, B, out);
}